// HybridLoss_36103495090314
// MI455X (gfx1250) — compile-verified
//
#include <hip/hip_runtime.h>
#include <hip/hip_bf16.h>

#define BATCH 8
#define HH 384
#define WW 384
#define HW (HH * WW)            // 147456
#define NTOT (BATCH * HW)       // 1179648

// K3 decomposition: 18 blocks/image * 8192 elems/block, 256 thr (8 waves),
// each wave owns 1024 contiguous elems -> 16 iters * 64 elems (2/lane).
#define BPI 18
#define CHUNK 8192
#define NBLK (BATCH * BPI)      // 144

typedef __attribute__((ext_vector_type(2))) float v2f;
typedef __attribute__((ext_vector_type(8))) float v8f;

#define F_INF 768.0f            // > any achievable 1D distance (H+W)

// ---------------------------------------------------------------------------
// K1: per-column vertical 1D EDT (two scans) for mask m and (1-m).
// One block per image (384 threads = 1 thread / column, coalesced row reads).
// Also computes per-image target sum and re-zeroes the maxabs atomic slot.
// Stores g^2 (squared vertical distance) for the envelope pass.
// ---------------------------------------------------------------------------
__global__ void k1_scan(const float* __restrict__ targ,
                        float* __restrict__ gf, float* __restrict__ gb,
                        float* __restrict__ s_img, unsigned* __restrict__ maxabs) {
    const int b = blockIdx.x;
    const int j = threadIdx.x;          // column
    if (j == 0) maxabs[b] = 0u;         // deterministic re-init each call
    const float* t = targ + b * HW;
    float* gfi = gf + b * HW;
    float* gbi = gb + b * HW;

    float cf = F_INF, cb = F_INF, cs = 0.0f;
    for (int i = 0; i < HH; ++i) {
        float v = t[i * WW + j];
        cs += v;
        cf = (v == 0.0f) ? 0.0f : cf + 1.0f;   // dist to nearest zero of m
        cb = (v == 0.0f) ? cb + 1.0f : 0.0f;   // dist to nearest zero of (1-m)
        gfi[i * WW + j] = cf;
        gbi[i * WW + j] = cb;
    }
    cf = F_INF; cb = F_INF;
    for (int i = HH - 1; i >= 0; --i) {
        float v = t[i * WW + j];
        cf = (v == 0.0f) ? 0.0f : cf + 1.0f;
        cb = (v == 0.0f) ? cb + 1.0f : 0.0f;
        int idx = i * WW + j;
        float a = fminf(gfi[idx], cf);  gfi[idx] = a * a;
        float c = fminf(gbi[idx], cb);  gbi[idx] = c * c;
    }

    // block reduce per-image target sum (384 = 3*128, fixed-order tree)
    __shared__ float red[WW];
    red[j] = cs;
    __syncthreads();
    for (int o = 192; o >= 3; o >>= 1) {
        if (j < o) red[j] += red[j + o];
        __syncthreads();
    }
    if (j == 0) s_img[b] = red[0] + red[1] + red[2];
}

// ---------------------------------------------------------------------------
// K2: lower envelope  d2[i,j] = min_j' ( g[i,j']^2 + (j-j')^2 )  for both
// masks, sd = sqrt(d2_bg) - sqrt(d2_fg), per-image max|sd| via u32 atomicMax.
// One block (128 thr) per (image,row). Candidate rows staged in LDS and read
// as float4 broadcasts (ds_load_b128, uniform address, conflict-free); each
// thread computes 3 outputs (j, j+128, j+256) in ONE pass over candidates so
// every LDS value is reused 6x (3 outputs x 2 masks) -> 6 independent
// fma/min chains per candidate, DS traffic off the critical path.
// ---------------------------------------------------------------------------
__global__ void k2_envelope(const float* __restrict__ gf, const float* __restrict__ gb,
                            float* __restrict__ sd, unsigned* __restrict__ maxabs) {
    const int row = blockIdx.x;              // 0 .. BATCH*HH-1
    const int b   = row / HH;
    const float* rf = gf + row * WW;
    const float* rb = gb + row * WW;

    __shared__ float4 hf4[WW / 4];
    __shared__ float4 hb4[WW / 4];
    {
        float* hf = (float*)hf4;
        float* hb = (float*)hb4;
        for (int j = threadIdx.x; j < WW; j += blockDim.x) {
            hf[j] = rf[j];
            hb[j] = rb[j];
        }
    }
    __syncthreads();

    const int   tid = threadIdx.x;           // outputs: tid, tid+128, tid+256
    float d2f0 = 3.4e38f, d2b0 = 3.4e38f;
    float d2f1 = 3.4e38f, d2b1 = 3.4e38f;
    float d2f2 = 3.4e38f, d2b2 = 3.4e38f;
    float base = (float)tid;                 // (jo0 - jp), decremented by 4/trip

#define CAND(fv, gv, dd)                                                   \
    {                                                                      \
        float c0 = (dd), c1 = (dd) + 128.0f, c2 = (dd) + 256.0f;           \
        d2f0 = fminf(d2f0, fmaf(c0, c0, (fv)));                            \
        d2b0 = fminf(d2b0, fmaf(c0, c0, (gv)));                            \
        d2f1 = fminf(d2f1, fmaf(c1, c1, (fv)));                            \
        d2b1 = fminf(d2b1, fmaf(c1, c1, (gv)));                            \
        d2f2 = fminf(d2f2, fmaf(c2, c2, (fv)));                            \
        d2b2 = fminf(d2b2, fmaf(c2, c2, (gv)));                            \
    }

    for (int q = 0; q < WW / 4; ++q) {
        float4 f = hf4[q];                   // ds_load_b128 broadcast
        float4 g = hb4[q];
        CAND(f.x, g.x, base);
        CAND(f.y, g.y, base - 1.0f);
        CAND(f.z, g.z, base - 2.0f);
        CAND(f.w, g.w, base - 3.0f);
        base -= 4.0f;
    }
#undef CAND

    float s0 = __builtin_sqrtf(d2b0) - __builtin_sqrtf(d2f0);
    float s1 = __builtin_sqrtf(d2b1) - __builtin_sqrtf(d2f1);
    float s2 = __builtin_sqrtf(d2b2) - __builtin_sqrtf(d2f2);
    sd[row * WW + tid]       = s0;
    sd[row * WW + tid + 128] = s1;
    sd[row * WW + tid + 256] = s2;
    float mloc = fmaxf(fmaxf(fabsf(s0), fabsf(s1)), fabsf(s2));

    __shared__ float red[128];
    red[tid] = mloc;
    __syncthreads();
    for (int o = 64; o > 0; o >>= 1) {
        if (tid < o) red[tid] = fmaxf(red[tid], red[tid + o]);
        __syncthreads();
    }
    if (tid == 0)
        atomicMax(maxabs + b, __float_as_uint(red[0]));   // >=0: uint order == float order
}

// ---------------------------------------------------------------------------
// K2b: per-image normalization scale, zeroed for trivial (all-0 / all-1) masks.
// ---------------------------------------------------------------------------
__global__ void k2b_scale(const float* __restrict__ s_img,
                          const unsigned* __restrict__ maxabs,
                          float* __restrict__ scale) {
    int b = threadIdx.x;
    if (b < BATCH) {
        float s  = s_img[b];
        float mx = __uint_as_float(maxabs[b]);
        bool trivial = (s == 0.0f) || (s == (float)HW);
        scale[b] = trivial ? 0.0f : 1.0f / (mx + 1e-6f);
    }
}

// ---------------------------------------------------------------------------
// K3: elementwise focal/sigmoid/boundary math + WMMA-based reductions.
// Five f32 accumulator tiles; V_WMMA_F32_16X16X4_F32 with A = ones(16x4)
// folds 64 new values per instruction into C (column sums), co-executing
// with the transcendental VALU work. Blocks never straddle an image.
// ---------------------------------------------------------------------------
__global__ void k3_reduce(const float* __restrict__ pred, const float* __restrict__ targ,
                          const float* __restrict__ sdist, const float* __restrict__ scale,
                          float* __restrict__ pF, float* __restrict__ pB,
                          float* __restrict__ pTp, float* __restrict__ pFp,
                          float* __restrict__ pFn) {
    const int blk  = blockIdx.x;
    const int b    = blk / BPI;
    const int base = b * HW + (blk % BPI) * CHUNK;
    const int wave = threadIdx.x >> 5;
    const int lane = threadIdx.x & 31;
    const int wbase = base + wave * 1024;
    const float sc = scale[b];

    v2f ones; ones[0] = 1.0f; ones[1] = 1.0f;
    v8f aF = {}, aB = {}, aTp = {}, aFp = {}, aFn = {};

    for (int it = 0; it < 16; ++it) {
        int e0 = wbase + it * 64 + lane;
        v2f bF, bB, bTp, bFp, bFn;
#pragma unroll
        for (int h = 0; h < 2; ++h) {
            int e = e0 + h * 32;
            float x = pred[e];
            float t = targ[e];
            // stable BCE-with-logits + focal (FA=0.5, FG=2)
            float sp  = fmaxf(x, 0.0f) + log1pf(__builtin_expf(-fabsf(x)));
            float bce = sp - x * t;
            float pt  = __builtin_expf(-bce);
            float om  = 1.0f - pt;
            float fo  = 0.5f * om * om * bce;
            float pr  = 1.0f / (1.0f + __builtin_expf(-x));   // sigmoid
            float bd  = pr * sdist[e] * sc;
            bF[h]  = fo;
            bB[h]  = bd;
            bTp[h] = pr * t;
            bFp[h] = pr * (1.0f - t);
            bFn[h] = (1.0f - pr) * t;
        }
        aF  = __builtin_amdgcn_wmma_f32_16x16x4_f32(false, ones, false, bF,  (short)0, aF,  false, false);
        aB  = __builtin_amdgcn_wmma_f32_16x16x4_f32(false, ones, false, bB,  (short)0, aB,  false, false);
        aTp = __builtin_amdgcn_wmma_f32_16x16x4_f32(false, ones, false, bTp, (short)0, aTp, false, false);
        aFp = __builtin_amdgcn_wmma_f32_16x16x4_f32(false, ones, false, bFp, (short)0, aFp, false, false);
        aFn = __builtin_amdgcn_wmma_f32_16x16x4_f32(false, ones, false, bFn, (short)0, aFn, false, false);
    }

    // C row 0 (vector comp 0): lane L holds column (L%16) sum; columns are
    // duplicated across the two half-waves -> full-lane sum is 2x the total.
    float sF = aF[0], sB = aB[0], sTp = aTp[0], sFp = aFp[0], sFn = aFn[0];
    for (int o = 16; o >= 1; o >>= 1) {
        sF  += __shfl_xor(sF,  o, 32);
        sB  += __shfl_xor(sB,  o, 32);
        sTp += __shfl_xor(sTp, o, 32);
        sFp += __shfl_xor(sFp, o, 32);
        sFn += __shfl_xor(sFn, o, 32);
    }

    __shared__ float part[8][5];
    if (lane == 0) {
        part[wave][0] = sF  * 0.5f;
        part[wave][1] = sB  * 0.5f;
        part[wave][2] = sTp * 0.5f;
        part[wave][3] = sFp * 0.5f;
        part[wave][4] = sFn * 0.5f;
    }
    __syncthreads();
    if (threadIdx.x == 0) {
        float tF = 0, tB = 0, tTp = 0, tFp = 0, tFn = 0;
        for (int w = 0; w < 8; ++w) {            // fixed order: deterministic
            tF += part[w][0]; tB += part[w][1]; tTp += part[w][2];
            tFp += part[w][3]; tFn += part[w][4];
        }
        pF[blk] = tF;  pB[blk] = tB;
        pTp[blk] = tTp; pFp[blk] = tFp; pFn[blk] = tFn;
    }
}

// ---------------------------------------------------------------------------
// K4: fold the 144 block partials in fixed order, assemble the 4 outputs.
// ---------------------------------------------------------------------------
__global__ void k4_final(const float* __restrict__ pF, const float* __restrict__ pB,
                         const float* __restrict__ pTp, const float* __restrict__ pFp,
                         const float* __restrict__ pFn, float* __restrict__ out) {
    if (threadIdx.x == 0 && blockIdx.x == 0) {
        float fsum = 0.0f, bsum = 0.0f;
        for (int i = 0; i < NBLK; ++i) { fsum += pF[i]; bsum += pB[i]; }
        float tl = 0.0f;
        for (int b = 0; b < BATCH; ++b) {
            float tp = 0, fp = 0, fn = 0;
            for (int k = 0; k < BPI; ++k) {
                int i = b * BPI + k;
                tp += pTp[i]; fp += pFp[i]; fn += pFn[i];
            }
            float tv = (tp + 1.0f) / (tp + 0.7f * fn + 0.3f * fp + 1.0f);
            tl += 1.0f - tv;
        }
        tl *= (1.0f / (float)BATCH);
        float fl = fsum * (1.0f / (float)NTOT);
        float bl = bsum * (1.0f / (float)NTOT);
        float loss = 0.35f * fl + 0.55f * tl + 0.10f * bl;
        out[0] = loss; out[1] = fl; out[2] = tl; out[3] = bl;
    }
}

// ---------------------------------------------------------------------------
extern "C" void kernel_launch(void* const* d_in, const int* in_sizes, int n_in,
                              void* d_out, int out_size, void* d_ws, size_t ws_size,
                              hipStream_t stream) {
    (void)in_sizes; (void)n_in; (void)out_size; (void)ws_size;
    const float* pred = (const float*)d_in[0];
    const float* targ = (const float*)d_in[1];
    float* out = (float*)d_out;
    float* ws  = (float*)d_ws;

    // workspace layout (floats)
    float*    gf     = ws;                       // [NTOT] squared vert EDT of m
    float*    gb     = ws + (size_t)NTOT;        // [NTOT] squared vert EDT of 1-m
    float*    sd     = ws + (size_t)2 * NTOT;    // [NTOT] signed distance (raw)
    float*    misc   = ws + (size_t)3 * NTOT;
    float*    s_img  = misc;                     // [8]
    unsigned* maxabs = (unsigned*)(misc + 8);    // [8] f32 bits, >=0
    float*    scale  = misc + 16;                // [8]
    float*    pF     = misc + 32;                // [144]
    float*    pB     = pF + NBLK;
    float*    pTp    = pB + NBLK;
    float*    pFp    = pTp + NBLK;
    float*    pFn    = pFp + NBLK;

    k1_scan    <<<BATCH,      WW, 0, stream>>>(targ, gf, gb, s_img, maxabs);
    k2_envelope<<<BATCH * HH, 128, 0, stream>>>(gf, gb, sd, maxabs);
    k2b_scale  <<<1,          32, 0, stream>>>(s_img, maxabs, scale);
    k3_reduce  <<<NBLK,      256, 0, stream>>>(pred, targ, sd, scale, pF, pB, pTp, pFp, pFn);
    k4_final   <<<1,           1, 0, stream>>>(pF, pB, pTp, pFp, pFn, out);
}